// VectorQuantizer_1692217114977
// MI455X (gfx1250) — compile-verified
//
#include <hip/hip_runtime.h>
#include <math.h>

#define N_EMBED   8192
#define EMBED_DIM 32
#define NQ        32768      // 32 batches * 32*32 spatial
#define HW        1024       // 32*32
#define N_TILES_Q 2048       // NQ/16
#define N_TILES_K 512        // N_EMBED/16
#define BETA      0.25f
#define EPS_NORM  1e-12f
#define EPS_ENT   1e-4f

#define CHUNK_TILES  8                         // code tiles staged per LDS chunk
#define CHUNK_FLOATS (CHUNK_TILES * 512)       // 4096 floats = 16 KB
#define N_CHUNKS     (N_TILES_K / CHUNK_TILES) // 64

typedef float v2f __attribute__((ext_vector_type(2)));
typedef float v8f __attribute__((ext_vector_type(8)));

// ---------------------------------------------------------------- normalize embedding rows
__global__ void norm_embed(const float* __restrict__ emb, float* __restrict__ ew) {
    int r = blockIdx.x * blockDim.x + threadIdx.x;      // 8192 rows
    const float* p = emb + (size_t)r * EMBED_DIM;
    float v[EMBED_DIM];
    float ss = 0.f;
#pragma unroll
    for (int c = 0; c < EMBED_DIM; ++c) { v[c] = p[c]; ss += v[c] * v[c]; }
    float rn = 1.f / fmaxf(sqrtf(ss), EPS_NORM);
    float* o = ew + (size_t)r * EMBED_DIM;
#pragma unroll
    for (int c = 0; c < EMBED_DIM; ++c) o[c] = v[c] * rn;
}

// ---------------------------------------------------------------- pre-swizzle ew into WMMA B-fragment layout
// B-frag (16x16x4 f32, wave32): lane L holds col n = L%16; VGPR v + half = row k = v + 2*(L/16) (+4*chunk)
// ewF[(ct*32 + lane)*16 + 2*c + v] = ew[ct*16 + (lane&15)][4*c + 2*(lane>>4) + v]
__global__ void swizzle_ew(const float* __restrict__ ew, float* __restrict__ ewF) {
    int tid  = blockIdx.x * blockDim.x + threadIdx.x;   // 512 tiles * 32 lanes
    int lane = tid & 31;
    int ct   = tid >> 5;
    int n    = ct * 16 + (lane & 15);
    int half = lane >> 4;
    const float* src = ew + (size_t)n * EMBED_DIM + 2 * half;
    float* dst = ewF + (size_t)tid * 16;
#pragma unroll
    for (int c = 0; c < 8; ++c) {
        dst[2 * c]     = src[4 * c];
        dst[2 * c + 1] = src[4 * c + 1];
    }
}

// ---------------------------------------------------------------- clear histogram + loss accumulator
__global__ void clear_ws(unsigned* __restrict__ hist, float* __restrict__ lossAcc) {
    int i = blockIdx.x * blockDim.x + threadIdx.x;
    if (i < N_EMBED) hist[i] = 0u;
    if (i == 0) *lossAcc = 0.f;
}

// ---------------------------------------------------------------- fused similarity + argmax (the 17.2 GFLOP core)
// One wave per 16-query tile; 8 waves/block share the codebook B-fragments through LDS
// (global traffic: 1 MB/block instead of 1 MB/wave -> 8x L2 traffic reduction).
__global__ void __launch_bounds__(256) argmax_kernel(const float* __restrict__ z,
                                                     const float* __restrict__ ewF,
                                                     int* __restrict__ idxbuf) {
    __shared__ float4 lbuf4[CHUNK_FLOATS / 4];          // 16 KB staging buffer

    int tid  = threadIdx.x;
    int lane = tid & 31;
    int half = lane >> 4;
    int tile = blockIdx.x * 8 + (tid >> 5);             // query tile, 0..2047

    // ---- A fragment: lane holds query row m = tile*16 + lane%16, dims k = 4c + 2*half + v
    // z is bchw: z[(m/1024)*32768 + k*1024 + (m%1024)]
    int mrow = tile * 16 + (lane & 15);
    const float* zb = z + (size_t)(mrow >> 10) * (EMBED_DIM * HW) + (mrow & (HW - 1));
    v2f a[8];
    float ssp = 0.f;
#pragma unroll
    for (int c = 0; c < 8; ++c) {
        float x0 = zb[(size_t)(4 * c + 2 * half + 0) * HW];
        float x1 = zb[(size_t)(4 * c + 2 * half + 1) * HW];
        a[c][0] = x0; a[c][1] = x1;
        ssp += x0 * x0 + x1 * x1;
    }
    // each half-wave holds half the dims of row m: combine across halves
    float ss = ssp + __shfl_xor(ssp, 16, 32);
    float rn = 1.f / fmaxf(sqrtf(ss), EPS_NORM);
#pragma unroll
    for (int c = 0; c < 8; ++c) { a[c][0] *= rn; a[c][1] *= rn; }

    float best[8];
    int   bidx[8];
#pragma unroll
    for (int j = 0; j < 8; ++j) { best[j] = -3.402823466e38f; bidx[j] = 0; }

    // ---- cooperative staging: thread tid owns 64B (4 x float4) of each 16 KB chunk
    const float4* gsrc  = (const float4*)ewF + (size_t)tid * 4;   // +1024 float4 per chunk
    float4*       lst   = lbuf4 + (size_t)tid * 4;
    const v2f*    lbase = (const v2f*)lbuf4 + (size_t)lane * 8;

    float4 st[4];
#pragma unroll
    for (int q = 0; q < 4; ++q) st[q] = gsrc[q];                  // prefetch chunk 0
    gsrc += 1024;

    for (int ch = 0; ch < N_CHUNKS; ++ch) {
#pragma unroll
        for (int q = 0; q < 4; ++q) lst[q] = st[q];               // ds_store (waits loadcnt here)
        if (ch + 1 < N_CHUNKS) {                                  // issue next chunk's loads early
#pragma unroll
            for (int q = 0; q < 4; ++q) st[q] = gsrc[q];
            gsrc += 1024;
        }
        __syncthreads();                                          // stores visible to all 8 waves

        const v2f* bp = lbase;
        for (int tt = 0; tt < CHUNK_TILES; ++tt) {
            v2f b[8];
#pragma unroll
            for (int c = 0; c < 8; ++c) b[c] = bp[c];             // ds_load_b128 x4
            bp += 256;

            v8f acc = {0.f, 0.f, 0.f, 0.f, 0.f, 0.f, 0.f, 0.f};
#pragma unroll
            for (int c = 0; c < 8; ++c)
                acc = __builtin_amdgcn_wmma_f32_16x16x4_f32(
                    false, a[c], false, b[c], (short)0, acc, false, false);

            // C/D layout: VGPR j, half -> query m = j + 8*half ; N = lane%16
            int nb = (ch * CHUNK_TILES + tt) * 16 + (lane & 15);
#pragma unroll
            for (int j = 0; j < 8; ++j) {
                bool gt = acc[j] > best[j];   // strict >, ascending order => first-occurrence ties
                best[j] = gt ? acc[j] : best[j];
                bidx[j] = gt ? nb     : bidx[j];
            }
        }
        __syncthreads();                                          // all waves done before overwrite
    }

    // reduce the 16 candidate lanes within each half (masks 1..8 never cross halves)
#pragma unroll
    for (int m = 1; m < 16; m <<= 1) {
#pragma unroll
        for (int j = 0; j < 8; ++j) {
            float ov = __shfl_xor(best[j], m, 32);
            int   oi = __shfl_xor(bidx[j], m, 32);
            bool take = (ov > best[j]) || (ov == best[j] && oi < bidx[j]);
            best[j] = take ? ov : best[j];
            bidx[j] = take ? oi : bidx[j];
        }
    }
    if ((lane & 15) == 0) {
#pragma unroll
        for (int j = 0; j < 8; ++j)
            idxbuf[tile * 16 + j + 8 * half] = bidx[j];
    }
}

// ---------------------------------------------------------------- codebook usage histogram
__global__ void hist_build(const int* __restrict__ idxbuf, unsigned* __restrict__ hist) {
    int i = blockIdx.x * blockDim.x + threadIdx.x;
    atomicAdd(&hist[idxbuf[i]], 1u);
}

// ---------------------------------------------------------------- gather z_q, write bchw output, loss partials, idx out
__global__ void gather_loss(const float* __restrict__ z,
                            const float* __restrict__ ew,
                            const int* __restrict__ idxbuf,
                            float* __restrict__ out_zq,
                            float* __restrict__ out_idx,
                            float* __restrict__ lossAcc) {
    int i = blockIdx.x * blockDim.x + threadIdx.x;   // 0..32767
    int code = idxbuf[i];
    const float* e  = ew + (size_t)code * EMBED_DIM;
    size_t boff = (size_t)(i >> 10) * (EMBED_DIM * HW) + (i & (HW - 1));
    const float* zp = z + boff;
    float* op = out_zq + boff;

    float zv[EMBED_DIM];
    float ss = 0.f;
#pragma unroll
    for (int c = 0; c < EMBED_DIM; ++c) { zv[c] = zp[(size_t)c * HW]; ss += zv[c] * zv[c]; }
    float rn = 1.f / fmaxf(sqrtf(ss), EPS_NORM);

    float d2 = 0.f;
#pragma unroll
    for (int c = 0; c < EMBED_DIM; ++c) {
        float q = e[c];                 // z_q = l2norm(embedding[idx]) == ew row
        float d = q - zv[c] * rn;
        d2 += d * d;
        op[(size_t)c * HW] = q;         // straight-through: zn + sg(zq - zn) == zq
    }
    out_idx[i] = (float)code;

#pragma unroll
    for (int m = 16; m >= 1; m >>= 1) d2 += __shfl_xor(d2, m, 32);
    if ((threadIdx.x & 31) == 0) atomicAdd(lossAcc, d2);
}

// ---------------------------------------------------------------- entropy + loss finalize (single block)
__global__ void finalize(const unsigned* __restrict__ hist,
                         const float* __restrict__ lossAcc,
                         float* __restrict__ out_scalars) {   // [0]=loss, [1]=entropy
    __shared__ float red[256];
    int t = threadIdx.x;
    float s = 0.f;
    for (int k = t; k < N_EMBED; k += 256) s += (float)hist[k] + EPS_ENT;
    red[t] = s; __syncthreads();
    for (int w = 128; w > 0; w >>= 1) { if (t < w) red[t] += red[t + w]; __syncthreads(); }
    float S = red[0];
    __syncthreads();
    float e = 0.f;
    for (int k = t; k < N_EMBED; k += 256) {
        float p = ((float)hist[k] + EPS_ENT) / S;
        e += p * logf(p);
    }
    red[t] = e; __syncthreads();
    for (int w = 128; w > 0; w >>= 1) { if (t < w) red[t] += red[t + w]; __syncthreads(); }
    if (t == 0) {
        out_scalars[0] = (1.f + BETA) * (*lossAcc) / (float)NQ;
        out_scalars[1] = -red[0];
    }
}

// ---------------------------------------------------------------- launch
extern "C" void kernel_launch(void* const* d_in, const int* in_sizes, int n_in,
                              void* d_out, int out_size, void* d_ws, size_t ws_size,
                              hipStream_t stream) {
    const float* z   = (const float*)d_in[0];   // [32,32,32,32] bchw
    const float* emb = (const float*)d_in[1];   // [8192,32]
    float* out = (float*)d_out;
    // d_out layout: zq[1048576] | loss | entropy | idx[32768]
    float* out_loss = out + (size_t)NQ * EMBED_DIM;
    float* out_idx  = out_loss + 2;

    char* ws = (char*)d_ws;
    float*    ew      = (float*)ws;                                           // 1 MB
    float*    ewF     = (float*)(ws + (size_t)N_EMBED * EMBED_DIM * 4);       // 1 MB
    int*      idxb    = (int*)(ws + (size_t)2 * N_EMBED * EMBED_DIM * 4);     // 128 KB
    unsigned* hist    = (unsigned*)((char*)idxb + (size_t)NQ * 4);            // 32 KB
    float*    lossAcc = (float*)((char*)hist + (size_t)N_EMBED * 4);          // 4 B

    norm_embed   <<<N_EMBED / 256,        256, 0, stream>>>(emb, ew);
    swizzle_ew   <<<(N_TILES_K * 32)/256, 256, 0, stream>>>(ew, ewF);
    clear_ws     <<<N_EMBED / 256,        256, 0, stream>>>(hist, lossAcc);
    argmax_kernel<<<N_TILES_Q / 8,        256, 0, stream>>>(z, ewF, idxb);    // 8 waves/block
    hist_build   <<<NQ / 256,             256, 0, stream>>>(idxb, hist);
    gather_loss  <<<NQ / 256,             256, 0, stream>>>(z, ew, idxb, out, out_idx, lossAcc);
    finalize     <<<1,                    256, 0, stream>>>(hist, lossAcc, out_loss);
}